// UnifiedAttentionTransformer_65481071409693
// MI455X (gfx1250) — compile-verified
//
#include <hip/hip_runtime.h>

typedef __attribute__((ext_vector_type(2))) float v2f;
typedef __attribute__((ext_vector_type(8))) float v8f;

#define E_DIM 1024
#define F_DIM 4096
#define KSPLIT1 8
#define KCHUNK1 (E_DIM / KSPLIT1)   // 128
#define KSPLIT2 16
#define KCHUNK2 (F_DIM / KSPLIT2)   // 256

// ---------------------------------------------------------------------------
// Kernel 1: qv via parallel prefix product.
// q_n = q + p - 2qp  ==>  (1-2q_n) = (1-2q)(1-2p), so with t_i = 1-2*p1[i],
// qv[i] = (1 - prod_{j<=i} t_j) / 2.  Hillis-Steele scan in LDS, 1024 threads.
// ---------------------------------------------------------------------------
__global__ __launch_bounds__(E_DIM) void qv_scan_kernel(
    const float* __restrict__ rot, float* __restrict__ qv) {
  __shared__ float s[E_DIM];
  const int i = threadIdx.x;
  const float a = 0.5f * rot[i];
  const float b = 0.5f * rot[E_DIM + i];
  const float u = sinf(b) * cosf(a);
  const float v = cosf(b) * sinf(a);
  const float p = u * u + v * v;
  s[i] = 1.0f - 2.0f * p;
  __syncthreads();
  for (int off = 1; off < E_DIM; off <<= 1) {
    float x = s[i];
    if (i >= off) x *= s[i - off];
    __syncthreads();
    s[i] = x;
    __syncthreads();
  }
  qv[i] = 0.5f * (1.0f - s[i]);
}

// ---------------------------------------------------------------------------
// Matvec via V_WMMA_F32_16X16X4_F32.
// A[m][k] = W[row m][k] * vec[k]  (vec folded into A), B = all ones
//   => every column of D holds the dot products; read lane's own D values.
// A layout (ISA 7.12.2): lane L holds A[M=L%16][K=2*(L>>4)+j] in vgpr j.
// D layout: vgpr r on lane L = D[M = r + 8*(L>>4)][N = L%16].
// Each wave produces 16 output rows over one K-chunk (K-split for occupancy).
// ---------------------------------------------------------------------------
__global__ __launch_bounds__(256) void ffn1_wmma_kernel(
    const float* __restrict__ W,       // (F_DIM, E_DIM) row-major
    const float* __restrict__ qv,      // (E_DIM)
    float* __restrict__ h_part) {      // (KSPLIT1, F_DIM) partial dots
  __shared__ float sQ[KCHUNK1];
  const int chunk = blockIdx.x & (KSPLIT1 - 1);
  const int tgrp  = blockIdx.x >> 3;          // 32 tile-groups
  const int wave  = threadIdx.x >> 5;
  const int lane  = threadIdx.x & 31;
  const int m     = lane & 15;
  const int kh    = lane >> 4;                // 0 or 1
  const int ebase = chunk * KCHUNK1;

  if (threadIdx.x < KCHUNK1) sQ[threadIdx.x] = qv[ebase + threadIdx.x];
  __syncthreads();

  const int fbase = (tgrp * 8 + wave) * 16;
  const float* row = W + (size_t)(fbase + m) * E_DIM + ebase;

  v2f ones = {1.0f, 1.0f};
  v8f acc = {0.f, 0.f, 0.f, 0.f, 0.f, 0.f, 0.f, 0.f};
  for (int e = 0; e < KCHUNK1; e += 4) {
    if ((e & 63) == 0) __builtin_prefetch(row + e + 256, 0, 1);  // global_prefetch_b8
    const int k = e + 2 * kh;
    v2f w = *(const v2f*)(row + k);
    v2f aop;
    aop[0] = w[0] * sQ[k];
    aop[1] = w[1] * sQ[k + 1];
    acc = __builtin_amdgcn_wmma_f32_16x16x4_f32(
        false, aop, false, ones, (short)0, acc, false, false);
  }
  if (m == 0) {  // lanes 0 and 16 write rows [0..7] and [8..15]
    float* dst = h_part + (size_t)chunk * F_DIM + fbase + 8 * kh;
#pragma unroll
    for (int r = 0; r < 8; ++r) dst[r] = acc[r];
  }
}

__global__ __launch_bounds__(256) void ffn2_wmma_kernel(
    const float* __restrict__ W2,      // (E_DIM, F_DIM) row-major
    const float* __restrict__ h_part,  // (KSPLIT1, F_DIM)
    const float* __restrict__ b1,      // (F_DIM)
    float* __restrict__ x_part) {      // (KSPLIT2, E_DIM)
  __shared__ float sH[KCHUNK2];
  const int chunk = blockIdx.x & (KSPLIT2 - 1);
  const int tgrp  = blockIdx.x >> 4;          // 8 tile-groups
  const int wave  = threadIdx.x >> 5;
  const int lane  = threadIdx.x & 31;
  const int m     = lane & 15;
  const int kh    = lane >> 4;
  const int fbase = chunk * KCHUNK2;

  // Reduce ffn1 K-partials, add bias, relu -> staged h chunk in LDS
  for (int i = threadIdx.x; i < KCHUNK2; i += 256) {
    const int f = fbase + i;
    float hv = b1[f];
#pragma unroll
    for (int s = 0; s < KSPLIT1; ++s) hv += h_part[(size_t)s * F_DIM + f];
    sH[i] = fmaxf(hv, 0.0f);
  }
  __syncthreads();

  const int ebase = (tgrp * 8 + wave) * 16;
  const float* row = W2 + (size_t)(ebase + m) * F_DIM + fbase;

  v2f ones = {1.0f, 1.0f};
  v8f acc = {0.f, 0.f, 0.f, 0.f, 0.f, 0.f, 0.f, 0.f};
  for (int f = 0; f < KCHUNK2; f += 4) {
    if ((f & 63) == 0) __builtin_prefetch(row + f + 256, 0, 1);
    const int k = f + 2 * kh;
    v2f w = *(const v2f*)(row + k);
    v2f aop;
    aop[0] = w[0] * sH[k];
    aop[1] = w[1] * sH[k + 1];
    acc = __builtin_amdgcn_wmma_f32_16x16x4_f32(
        false, aop, false, ones, (short)0, acc, false, false);
  }
  if (m == 0) {
    float* dst = x_part + (size_t)chunk * E_DIM + ebase + 8 * kh;
#pragma unroll
    for (int r = 0; r < 8; ++r) dst[r] = acc[r];
  }
}

// ---------------------------------------------------------------------------
// Kernel 4: x[e] = sum of K-partials + b2;  out = dot(clf_w, x) + clf_b,
// replicated B=8 times (mean over S is identity since x is constant in s).
// ---------------------------------------------------------------------------
__global__ __launch_bounds__(E_DIM) void final_kernel(
    const float* __restrict__ x_part, const float* __restrict__ b2,
    const float* __restrict__ clf_w, const float* __restrict__ clf_b,
    float* __restrict__ out) {
  __shared__ float s[E_DIM];
  const int t = threadIdx.x;
  float xe = b2[t];
#pragma unroll
  for (int p = 0; p < KSPLIT2; ++p) xe += x_part[(size_t)p * E_DIM + t];
  s[t] = clf_w[t] * xe;
  __syncthreads();
  for (int off = E_DIM / 2; off > 0; off >>= 1) {
    if (t < off) s[t] += s[t + off];
    __syncthreads();
  }
  if (t < 8) out[t] = s[0] + clf_b[0];
}

extern "C" void kernel_launch(void* const* d_in, const int* in_sizes, int n_in,
                              void* d_out, int out_size, void* d_ws, size_t ws_size,
                              hipStream_t stream) {
  (void)in_sizes; (void)n_in; (void)out_size; (void)ws_size;
  // setup_inputs order:
  // 0 input_ids (dead), 1 rotation_params, 2 entangle_params (dead),
  // 3 emb_table (dead), 4 ffn1_w, 5 ffn1_b, 6 ffn2_w, 7 ffn2_b, 8 clf_w, 9 clf_b
  const float* rot   = (const float*)d_in[1];
  const float* ffn1w = (const float*)d_in[4];
  const float* ffn1b = (const float*)d_in[5];
  const float* ffn2w = (const float*)d_in[6];
  const float* ffn2b = (const float*)d_in[7];
  const float* clfw  = (const float*)d_in[8];
  const float* clfb  = (const float*)d_in[9];

  float* ws     = (float*)d_ws;
  float* qv     = ws;                                // 1024 f
  float* h_part = ws + E_DIM;                        // 8*4096 f
  float* x_part = ws + E_DIM + KSPLIT1 * F_DIM;      // 16*1024 f (total ~196 KB)
  float* out    = (float*)d_out;

  qv_scan_kernel<<<1, E_DIM, 0, stream>>>(rot, qv);
  // 256 row-tiles * KSPLIT1 chunks / 8 waves-per-block = 256 blocks
  ffn1_wmma_kernel<<<(F_DIM / 16 / 8) * KSPLIT1, 256, 0, stream>>>(ffn1w, qv, h_part);
  // 64 row-tiles * KSPLIT2 chunks / 8 waves-per-block = 128 blocks
  ffn2_wmma_kernel<<<(E_DIM / 16 / 8) * KSPLIT2, 256, 0, stream>>>(ffn2w, h_part, ffn1b, x_part);
  final_kernel<<<1, E_DIM, 0, stream>>>(x_part, ffn2b, clfw, clfb, out);
}